// NeuralODE_42193758715947
// MI455X (gfx1250) — compile-verified
//
#include <hip/hip_runtime.h>
#include <math.h>

// ---------------------------------------------------------------------------
// NeuralODE RK4 on MI455X (gfx1250), full fp32 via V_WMMA_F32_16X16X4_F32,
// hardware tanh via V_TANH_F32.
//
// One wave32 integrates a 16-row batch tile through all 2000 RK4 steps.
//   BATCH=8192 -> 512 tiles -> 512 single-wave blocks (max WGP spread;
//   ~29 KB LDS/block lets several blocks stack per WGP for latency hiding).
//
// Layouts (CDNA5 ISA 7.12.2, wave32):
//   A 16x4 f32 : v[0] = lanes0-15 -> (M=lane, K=0), lanes16-31 -> (M=lane-16, K=2)
//                v[1] = K=1 / K=3
//   B 4x16 f32 : v[0] = lanes0-15 -> (K=0, N=lane), lanes16-31 -> (K=2, N=lane-16)
//                v[1] = K=1 / K=3
//   C/D 16x16  : VGPR j, lane l -> (M = j + 8*(l>=16), N = l%16)
// ---------------------------------------------------------------------------

typedef __attribute__((ext_vector_type(2))) float v2f;
typedef __attribute__((ext_vector_type(8))) float v8f;

#define DTF        0.01f
#define NSTEPS     2000
#define BATCH      8192
#define TRAJ       (NSTEPS + 1)
#define HSTR       66           // hidden-buffer row stride (64 + 2: kills bank conflicts)

__device__ __forceinline__ v8f wmma_f32_k4(v2f a, v2f b, v8f c) {
  return __builtin_amdgcn_wmma_f32_16x16x4_f32(
      /*neg_a=*/false, a, /*neg_b=*/false, b,
      /*c_mod=*/(short)0, c, /*reuse_a=*/false, /*reuse_b=*/false);
}

__device__ __forceinline__ void lds_wait() {
  asm volatile("s_wait_dscnt 0x0" ::: "memory");
}

// Branch-free tanh. Preferred: gfx1250 hardware V_TANH_F32 (single TRANS op).
// Fallback: exp2/rcp formulation -- still zero EXEC divergence.
__device__ __forceinline__ float fast_tanh(float x) {
#if __has_builtin(__builtin_amdgcn_tanhf)
  return __builtin_amdgcn_tanhf(x);
#else
  const float ax = __builtin_fabsf(x);
  const float e  = __builtin_amdgcn_exp2f(ax * -2.885390081777927f); // e^(-2|x|)
  const float r  = __builtin_amdgcn_rcpf(1.0f + e);
  const float t  = __builtin_fmaf(-2.0f * e, r, 1.0f);               // (1-e)/(1+e)
  return __builtin_copysignf(t, x);
#endif
}

// One MLP evaluation for a 16-row tile: (yx,yy) -> (kx,ky).
// All lanes hold valid y (hi lanes mirror lo lanes); the A-frag select zeroes
// hi lanes (they carry the K=2/3 pad). All WMMAs are in uniform control flow.
__device__ __forceinline__ void net_eval(
    const float* __restrict__ w1f, const float* __restrict__ w2f,
    const float* __restrict__ w3f,
    float* __restrict__ hb1, float* __restrict__ hb2, float* __restrict__ ytmp,
    const float* __restrict__ b1v, const float* __restrict__ b2v,
    float b3x, float b3y,
    float yx, float yy, float& kx, float& ky)
{
  const int l   = threadIdx.x & 31;
  const int hi  = l >> 4;        // 0: lanes 0-15, 1: lanes 16-31
  const int n16 = l & 15;

  // ---- layer 1: h1 = tanh(y @ W1 + b1), y is 16x2 zero-padded to 16x4 ----
  v2f a1;
  a1.x = hi ? 0.0f : yx;         // K=0 (lo) / K=2 pad (hi)
  a1.y = hi ? 0.0f : yy;         // K=1 (lo) / K=3 pad (hi)
#pragma unroll
  for (int nt = 0; nt < 4; ++nt) {
    v2f b = *(const v2f*)&w1f[(nt * 32 + l) * 2];
    v8f c;
#pragma unroll
    for (int j = 0; j < 8; ++j) c[j] = b1v[nt];        // bias preloaded into C
    c = wmma_f32_k4(a1, b, c);
#pragma unroll
    for (int j = 0; j < 8; ++j)
      hb1[(j + 8 * hi) * HSTR + nt * 16 + n16] = fast_tanh(c[j]);
  }
  lds_wait();

  // ---- layer 2: h2 = tanh(h1 @ W2 + b2): 16 K-chunks x 4 N-tiles ----
#pragma unroll
  for (int nt = 0; nt < 4; ++nt) {
    v8f acc;
#pragma unroll
    for (int j = 0; j < 8; ++j) acc[j] = b2v[nt];
#pragma unroll
    for (int kc = 0; kc < 16; ++kc) {
      v2f a = *(const v2f*)&hb1[n16 * HSTR + 4 * kc + 2 * hi];     // A-frag
      v2f b = *(const v2f*)&w2f[((kc * 4 + nt) * 32 + l) * 2];     // B-frag
      acc = wmma_f32_k4(a, b, acc);
    }
#pragma unroll
    for (int j = 0; j < 8; ++j)
      hb2[(j + 8 * hi) * HSTR + nt * 16 + n16] = fast_tanh(acc[j]);
  }
  lds_wait();

  // ---- layer 3: k = h2 @ W3 + b3, N padded 2 -> 16 (cols >=2 zero) ----
  v8f acc = {};
#pragma unroll
  for (int kc = 0; kc < 16; ++kc) {
    v2f a = *(const v2f*)&hb2[n16 * HSTR + 4 * kc + 2 * hi];
    v2f b = *(const v2f*)&w3f[(kc * 32 + l) * 2];
    acc = wmma_f32_k4(a, b, acc);
  }
  // D layout -> per-lane (row = n16) float2 via tiny LDS transpose
  if (n16 < 2) {
#pragma unroll
    for (int j = 0; j < 8; ++j)
      ytmp[(j + 8 * hi) * 2 + n16] = acc[j];
  }
  lds_wait();
  v2f k = *(const v2f*)&ytmp[n16 * 2];
  kx = k.x + b3x;
  ky = k.y + b3y;
}

__global__ __launch_bounds__(32)
void node_rk4_kernel(const float* __restrict__ y0,
                     const float* __restrict__ W1, const float* __restrict__ b1,
                     const float* __restrict__ W2, const float* __restrict__ b2,
                     const float* __restrict__ W3, const float* __restrict__ b3,
                     float* __restrict__ out)
{
  // LDS (~29.4 KB): fragment-packed weights + activation round-trip buffers
  __shared__ float w1f[4 * 32 * 2];          //  1 KB
  __shared__ float w2f[64 * 32 * 2];         // 16 KB
  __shared__ float w3f[16 * 32 * 2];         //  4 KB
  __shared__ float hb1[16 * HSTR];           //  ~4.2 KB
  __shared__ float hb2[16 * HSTR];           //  ~4.2 KB
  __shared__ float ytmp[32];                 //  128 B

  const int l    = threadIdx.x & 31;
  const int hi   = l >> 4;
  const int n16  = l & 15;

  // ---- pack weights into WMMA B-fragment order (single wave per block) ----
#pragma unroll
  for (int f = 0; f < 4; ++f) {              // W1: 2x64, K padded to 4
    int n = f * 16 + n16;
    w1f[(f * 32 + l) * 2 + 0] = hi ? 0.0f : W1[n];        // K = 0 / 2(pad)
    w1f[(f * 32 + l) * 2 + 1] = hi ? 0.0f : W1[64 + n];   // K = 1 / 3(pad)
  }
  for (int f = 0; f < 64; ++f) {             // W2: 64x64, frag (kc, nt)
    int kc = f >> 2, nt = f & 3;
    int k0 = 4 * kc + 2 * hi;
    int n  = nt * 16 + n16;
    w2f[(f * 32 + l) * 2 + 0] = W2[k0 * 64 + n];
    w2f[(f * 32 + l) * 2 + 1] = W2[(k0 + 1) * 64 + n];
  }
#pragma unroll
  for (int f = 0; f < 16; ++f) {             // W3: 64x2, N padded to 16
    int k0 = 4 * f + 2 * hi;
    w3f[(f * 32 + l) * 2 + 0] = (n16 < 2) ? W3[k0 * 2 + n16] : 0.0f;
    w3f[(f * 32 + l) * 2 + 1] = (n16 < 2) ? W3[(k0 + 1) * 2 + n16] : 0.0f;
  }
  __syncthreads();   // single-wave WG: lowers to waits (barrier -> S_NOP)
  lds_wait();

  // ---- per-lane bias values in registers (depend only on N = lane%16) ----
  float b1v[4], b2v[4];
#pragma unroll
  for (int nt = 0; nt < 4; ++nt) {
    b1v[nt] = b1[nt * 16 + n16];
    b2v[nt] = b2[nt * 16 + n16];
  }
  const float b3x = b3[0], b3y = b3[1];

  // ---- load this block's 16-row state tile (hi lanes mirror lo lanes) ----
  const int tile = blockIdx.x;               // 0..511
  const int base = tile * 16;
  v2f y = *(const v2f*)&y0[(base + n16) * 2];
  float yx = y.x, yy = y.y;

  // t = 0 trajectory sample (streamed, never re-read -> non-temporal)
  if (l < 16) {
    v2f o; o.x = yx; o.y = yy;
    __builtin_nontemporal_store(o, (v2f*)&out[(size_t)(base + l) * (TRAJ * 2)]);
  }

  // ---- 2000 RK4 steps ----
  for (int t = 1; t <= NSTEPS; ++t) {
    float k1x, k1y, k2x, k2y, k3x, k3y, k4x, k4y;
    net_eval(w1f, w2f, w3f, hb1, hb2, ytmp, b1v, b2v, b3x, b3y,
             yx, yy, k1x, k1y);
    net_eval(w1f, w2f, w3f, hb1, hb2, ytmp, b1v, b2v, b3x, b3y,
             yx + 0.5f * DTF * k1x, yy + 0.5f * DTF * k1y, k2x, k2y);
    net_eval(w1f, w2f, w3f, hb1, hb2, ytmp, b1v, b2v, b3x, b3y,
             yx + 0.5f * DTF * k2x, yy + 0.5f * DTF * k2y, k3x, k3y);
    net_eval(w1f, w2f, w3f, hb1, hb2, ytmp, b1v, b2v, b3x, b3y,
             yx + DTF * k3x, yy + DTF * k3y, k4x, k4y);
    yx += (DTF / 6.0f) * (k1x + 2.0f * k2x + 2.0f * k3x + k4x);
    yy += (DTF / 6.0f) * (k1y + 2.0f * k2y + 2.0f * k3y + k4y);

    if (l < 16) {
      v2f o; o.x = yx; o.y = yy;
      __builtin_nontemporal_store(
          o, (v2f*)&out[(size_t)(base + l) * (TRAJ * 2) + (size_t)t * 2]);
    }
  }
}

extern "C" void kernel_launch(void* const* d_in, const int* in_sizes, int n_in,
                              void* d_out, int out_size, void* d_ws, size_t ws_size,
                              hipStream_t stream) {
  const float* y0 = (const float*)d_in[0];
  const float* W1 = (const float*)d_in[1];
  const float* b1 = (const float*)d_in[2];
  const float* W2 = (const float*)d_in[3];
  const float* b2 = (const float*)d_in[4];
  const float* W3 = (const float*)d_in[5];
  const float* b3 = (const float*)d_in[6];

  const int blocks = BATCH / 16;             // 512 single-wave blocks
  node_rk4_kernel<<<blocks, 32, 0, stream>>>(y0, W1, b1, W2, b2, W3, b3,
                                             (float*)d_out);
}